// cfrm_new_88356067213422
// MI455X (gfx1250) — compile-verified
//
#include <hip/hip_runtime.h>
#include <math.h>

typedef __attribute__((ext_vector_type(2))) float v2f;
typedef __attribute__((ext_vector_type(8))) float v8f;
typedef __attribute__((ext_vector_type(4))) uint32_t u32x4;
typedef __attribute__((ext_vector_type(8))) uint32_t u32x8;

#define NF 100

// ---------------- TDM 2D tile load: global -> LDS ----------------
// Descriptor per CDNA5 ISA 8.3/8.4 (group0 128b, group1 256b), 2D tensor so
// VADDR2/VADDR3 groups are omitted. data_size=2 (4 bytes). type=2 ("image").
__device__ __forceinline__ void tdm_load_2d(uint32_t lds_off, uint64_t ga,
                                            uint32_t td0, uint32_t td1,
                                            uint32_t tile0, uint32_t tile1,
                                            uint32_t stride0, uint32_t padcfg) {
  u32x4 g0;
  g0[0] = 1u;                                                  // count=1
  g0[1] = lds_off;                                             // lds_addr
  g0[2] = (uint32_t)ga;                                        // global_addr lo
  g0[3] = ((uint32_t)(ga >> 32) & 0x01FFFFFFu) | (2u << 30);   // addr hi | type=2
  u32x8 g1;
  g1[0] = 0x00020000u | padcfg;  // wg_mask=0, data_size=4B, pad cfg
  g1[1] = td0 << 16;             // tensor_dim0[15:0]
  g1[2] = (td0 >> 16) | (td1 << 16);
  g1[3] = (td1 >> 16) | (tile0 << 16);
  g1[4] = tile1;                 // tile_dim1 | tile_dim2=0
  g1[5] = stride0;               // tensor_dim0_stride lo
  g1[6] = 0u;
  g1[7] = 0u;
  asm volatile("tensor_load_to_lds %0, %1" :: "s"(g0), "s"(g1) : "memory");
}

// ---------------- zero accumulators ----------------
__global__ void k_zero(double* acc, int n) {
  int i = blockIdx.x * blockDim.x + threadIdx.x;
  if (i < n) acc[i] = 0.0;
}

// ---------------- row softmax over 100 cols ----------------
__global__ void k_softmax(const float* __restrict__ C, float* __restrict__ S, int nrows) {
  int wave = threadIdx.x >> 5;
  int lane = threadIdx.x & 31;
  int row = blockIdx.x * (blockDim.x >> 5) + wave;
  if (row >= nrows) return;
  const float* cr = C + (size_t)row * NF;
  float v[4];
  float mx = -3.4e38f;
#pragma unroll
  for (int j = 0; j < 4; ++j) {
    int col = lane + 32 * j;
    v[j] = (col < NF) ? cr[col] : -3.4e38f;
    mx = fmaxf(mx, v[j]);
  }
  for (int off = 16; off; off >>= 1) mx = fmaxf(mx, __shfl_xor(mx, off, 32));
  float sum = 0.f;
#pragma unroll
  for (int j = 0; j < 4; ++j) {
    int col = lane + 32 * j;
    v[j] = (col < NF) ? __expf(v[j] - mx) : 0.f;
    sum += v[j];
  }
  for (int off = 16; off; off >>= 1) sum += __shfl_xor(sum, off, 32);
  float inv = 1.f / sum;
  float* sr = S + (size_t)row * NF;
#pragma unroll
  for (int j = 0; j < 4; ++j) {
    int col = lane + 32 * j;
    if (col < NF) sr[col] = v[j] * inv;
  }
}

// ---------------- T = S @ Amat : [rows,100] @ [100,100] ----------------
__global__ void k_gemm_small(const float* __restrict__ S, const float* __restrict__ Amat,
                             float* __restrict__ T) {
  __shared__ float srow[NF];
  int row = blockIdx.x;
  const float* sp = S + (size_t)row * NF;
  for (int i = threadIdx.x; i < NF; i += blockDim.x) srow[i] = sp[i];
  __syncthreads();
  int t = threadIdx.x;
  if (t < NF) {
    float acc = 0.f;
#pragma unroll 4
    for (int k = 0; k < NF; ++k) acc = fmaf(srow[k], Amat[k * NF + t], acc);
    T[(size_t)row * NF + t] = acc;
  }
}

// ---------------- fused reconstruction squared-error ----------------
// acc += sum over tile of (X - T·S^T - brow - bcol)^2
// block = 160 thr = 5 waves; M block 64 (4 subtiles/wave), N block 80 (16/wave)
// grid = (Ndim/80, 8000/64). T strip [64x100] and S strip [80x100] staged in
// LDS via two parallel TDM DMAs (wave0 / wave1, each waits its own TENSORcnt).
__global__ void k_loss_rec(const float* __restrict__ X,
                           const float* __restrict__ T,
                           const float* __restrict__ Smat,
                           const float* __restrict__ brow,
                           const float* __restrict__ bcol,
                           int Ndim, double* __restrict__ acc) {
  __shared__ float Ts[64 * NF];
  __shared__ float Ss[80 * NF];
  __shared__ float red[5];
  int wave = threadIdx.x >> 5;
  int lane = threadIdx.x & 31;
  int half = lane >> 4;
  int l = lane & 15;
  int m0 = blockIdx.y * 64;
  int n0b = blockIdx.x * 80;
  int n0 = n0b + wave * 16;

  if (wave == 0) {
    tdm_load_2d((uint32_t)(uintptr_t)&Ts[0],
                (uint64_t)(uintptr_t)(T + (size_t)m0 * NF),
                /*td0=*/NF, /*td1=*/8000, /*tile0=*/NF, /*tile1=*/64,
                /*stride0=*/NF, /*padcfg=*/0u);
    __builtin_amdgcn_s_wait_tensorcnt(0);
  } else if (wave == 1) {
    tdm_load_2d((uint32_t)(uintptr_t)&Ss[0],
                (uint64_t)(uintptr_t)(Smat + (size_t)n0b * NF),
                /*td0=*/NF, /*td1=*/(uint32_t)Ndim, /*tile0=*/NF, /*tile1=*/80,
                /*stride0=*/NF, /*padcfg=*/0u);
    __builtin_amdgcn_s_wait_tensorcnt(0);
  }
  __syncthreads();

  // B fragment: lane=N (local row wave*16+l); v0/v1 = K/K+1 (low) or K+2/K+3 (high)
  const float* bp = Ss + (wave * 16 + l) * NF + 2 * half;
  v8f c[4];
#pragma unroll
  for (int i = 0; i < 4; ++i) c[i] = (v8f){0.f,0.f,0.f,0.f,0.f,0.f,0.f,0.f};

#pragma unroll 5
  for (int k = 0; k < NF; k += 4) {
    v2f b;
    b.x = bp[k];
    b.y = bp[k + 1];
    int ka = k + 2 * half;
#pragma unroll
    for (int i = 0; i < 4; ++i) {
      v2f a;
      a.x = Ts[(i * 16 + l) * NF + ka];
      a.y = Ts[(i * 16 + l) * NF + ka + 1];
      c[i] = __builtin_amdgcn_wmma_f32_16x16x4_f32(false, a, false, b, (short)0, c[i], false, false);
    }
  }

  int colg = n0 + l;
  float bn = brow[colg];
  float part = 0.f;
#pragma unroll
  for (int i = 0; i < 4; ++i) {
    int mb = m0 + i * 16 + 8 * half;
#pragma unroll
    for (int r = 0; r < 8; ++r) {
      float x = X[(size_t)(mb + r) * Ndim + colg];
      float d = x - c[i][r] - bn - bcol[mb + r];
      part = fmaf(d, d, part);
    }
  }
  for (int off = 16; off; off >>= 1) part += __shfl_down(part, off, 32);
  if (lane == 0) red[wave] = part;
  __syncthreads();
  if (threadIdx.x == 0) {
    float s = red[0] + red[1] + red[2] + red[3] + red[4];
    atomicAdd(acc, (double)s);
  }
}

// ---------------- loss4: ASr = A @ Sr tiles, reduced in place ----------------
// A: [8000,20000]; Sr: [20000,100]; Sg: [8000,100]
// block = 224 thr = 7 waves (N=0..111, cols>=100 masked); M block 32 (2 subtiles).
// A chunks [32 x 160] double-buffered in LDS via TDM with HW padding:
// pad 1 DWORD every 32 DWORDs -> LDS row stride 165 (gcd(165,64)=1, conflict-free).
// acc[0]+=<Sg,ASr>, acc[1]+=||ASr||^2, acc[2]+=||Sg||^2
#define KC4 160
#define LROW 165
#define PADCFG ((1u << 20) | (4u << 22))  // pad_enable, interval=32 DW, amount=1 DW
__global__ void k_loss4(const float* __restrict__ A,
                        const float* __restrict__ Sr,
                        const float* __restrict__ Sg,
                        int Kdim, double* __restrict__ acc) {
  __shared__ float Asb[2][32 * LROW];
  __shared__ float red[7 * 3];
  int wave = threadIdx.x >> 5;
  int lane = threadIdx.x & 31;
  int half = lane >> 4;
  int l = lane & 15;
  int m0 = blockIdx.x * 32;
  int n0 = wave * 16;
  int col = n0 + l;
  bool cvalid = (col < NF);

  uint32_t lds0 = (uint32_t)(uintptr_t)&Asb[0][0];
  uint32_t lds1 = (uint32_t)(uintptr_t)&Asb[1][0];
  const uint64_t arow = (uint64_t)(uintptr_t)(A + (size_t)m0 * Kdim);

  if (wave == 0) {
    tdm_load_2d(lds0, arow, /*td0=*/20000u, /*td1=*/8000u,
                /*tile0=*/KC4, /*tile1=*/32, /*stride0=*/20000u, PADCFG);
    __builtin_amdgcn_s_wait_tensorcnt(0);
  }
  __syncthreads();

  v8f c[2];
  c[0] = (v8f){0.f,0.f,0.f,0.f,0.f,0.f,0.f,0.f};
  c[1] = (v8f){0.f,0.f,0.f,0.f,0.f,0.f,0.f,0.f};

  int buf = 0;
  for (int kc = 0; kc < Kdim; kc += KC4) {
    if (wave == 0 && kc + KC4 < Kdim) {
      tdm_load_2d(buf ? lds0 : lds1, arow + (uint64_t)(kc + KC4) * 4u,
                  20000u, 8000u, KC4, 32, 20000u, PADCFG);
    }
    const float* Ab = buf ? &Asb[1][0] : &Asb[0][0];
#pragma unroll 4
    for (int kk = 0; kk < KC4; kk += 4) {
      int ka = kk + 2 * half;
      int kl = ka + (ka >> 5);  // account for HW pad: +1 DW per 32 DW
      int kg = kc + ka;
      v2f b;
      b.x = cvalid ? Sr[(size_t)kg * NF + col] : 0.f;
      b.y = cvalid ? Sr[(size_t)(kg + 1) * NF + col] : 0.f;
#pragma unroll
      for (int i = 0; i < 2; ++i) {
        v2f a;
        a.x = Ab[(i * 16 + l) * LROW + kl];
        a.y = Ab[(i * 16 + l) * LROW + kl + 1];
        c[i] = __builtin_amdgcn_wmma_f32_16x16x4_f32(false, a, false, b, (short)0, c[i], false, false);
      }
    }
    if (wave == 0) __builtin_amdgcn_s_wait_tensorcnt(0);
    __syncthreads();
    buf ^= 1;
  }

  float dot = 0.f, nA = 0.f, nS = 0.f;
  if (cvalid) {
#pragma unroll
    for (int i = 0; i < 2; ++i) {
      int mb = m0 + i * 16 + 8 * half;
#pragma unroll
      for (int r = 0; r < 8; ++r) {
        float asr = c[i][r];
        float sg = Sg[(size_t)(mb + r) * NF + col];
        dot = fmaf(asr, sg, dot);
        nA = fmaf(asr, asr, nA);
        nS = fmaf(sg, sg, nS);
      }
    }
  }
  for (int off = 16; off; off >>= 1) {
    dot += __shfl_down(dot, off, 32);
    nA += __shfl_down(nA, off, 32);
    nS += __shfl_down(nS, off, 32);
  }
  if (lane == 0) { red[wave * 3 + 0] = dot; red[wave * 3 + 1] = nA; red[wave * 3 + 2] = nS; }
  __syncthreads();
  if (threadIdx.x == 0) {
    float d = 0.f, a2 = 0.f, s2 = 0.f;
    for (int w = 0; w < 7; ++w) { d += red[w * 3]; a2 += red[w * 3 + 1]; s2 += red[w * 3 + 2]; }
    atomicAdd(acc + 0, (double)d);
    atomicAdd(acc + 1, (double)a2);
    atomicAdd(acc + 2, (double)s2);
  }
}

// ---------------- loss3 terms over 100x100 ----------------
__global__ void k_loss3(const float* __restrict__ Ag, const float* __restrict__ Ar,
                        double* __restrict__ acc) {
  float d = 0.f, r2 = 0.f, g2 = 0.f;
  for (int i = threadIdx.x; i < NF * NF; i += blockDim.x) {
    float g = Ag[i], r = Ar[i];
    d = fmaf(r, g, d);
    r2 = fmaf(r, r, r2);
    g2 = fmaf(g, g, g2);
  }
  for (int off = 16; off; off >>= 1) {
    d += __shfl_down(d, off, 32);
    r2 += __shfl_down(r2, off, 32);
    g2 += __shfl_down(g2, off, 32);
  }
  if ((threadIdx.x & 31) == 0) {
    atomicAdd(acc + 0, (double)d);
    atomicAdd(acc + 1, (double)r2);
    atomicAdd(acc + 2, (double)g2);
  }
}

// ---------------- finalize ----------------
// acc: [0]=sum1 [1]=sum2 [2]=dotSgASr [3]=||ASr||^2 [4]=||Sg||^2 [5]=<Ar,Ag> [6]=||Ar||^2 [7]=||Ag||^2
__global__ void k_final(const double* __restrict__ acc, const float* __restrict__ alpha,
                        float* __restrict__ out) {
  if (threadIdx.x == 0 && blockIdx.x == 0) {
    double loss1 = acc[0] / (8000.0 * 8000.0);
    double loss2 = acc[1] / (8000.0 * 20000.0);
    double loss3 = -acc[5] / sqrt(acc[6]) / sqrt(acc[7]);
    double loss4 = -acc[2] / sqrt(acc[4]) / sqrt(acc[3]);
    double l1 = (double)alpha[0] * loss1;
    double l2 = (double)alpha[1] * loss2;
    double l3 = (double)alpha[2] * loss3;
    double l4 = (double)alpha[3] * loss4;
    out[0] = (float)(l1 + l2 + l3 + l4);
    out[1] = (float)l1;
    out[2] = (float)l2;
    out[3] = (float)l3;
    out[4] = (float)l4;
  }
}

extern "C" void kernel_launch(void* const* d_in, const int* in_sizes, int n_in,
                              void* d_out, int out_size, void* d_ws, size_t ws_size,
                              hipStream_t stream) {
  const float* G   = (const float*)d_in[0];   // [8000,8000]
  const float* R   = (const float*)d_in[1];   // [8000,20000]
  const float* A   = (const float*)d_in[2];   // [8000,20000]
  const float* C1  = (const float*)d_in[3];   // [8000,100]
  const float* C2  = (const float*)d_in[4];   // [8000,100]
  const float* Cg  = (const float*)d_in[5];   // [8000,100]
  const float* Cr  = (const float*)d_in[6];   // [20000,100]
  const float* Agm = (const float*)d_in[7];   // [100,100]
  const float* Arm = (const float*)d_in[8];   // [100,100]
  const float* b_g = (const float*)d_in[9];   // [1,8000]
  const float* b_r = (const float*)d_in[10];  // [1,20000]
  const float* b_1 = (const float*)d_in[11];  // [8000,1]
  const float* b_2 = (const float*)d_in[12];  // [8000,1]
  const float* alpha = (const float*)d_in[13];
  float* out = (float*)d_out;

  float* S1 = (float*)d_ws;
  float* S2 = S1 + 8000 * NF;
  float* Sg = S2 + 8000 * NF;
  float* Sr = Sg + 8000 * NF;
  float* T1 = Sr + 20000 * NF;
  float* T2 = T1 + 8000 * NF;
  double* accs = (double*)(T2 + 8000 * NF);  // byte offset 24,000,000 (8-aligned)

  k_zero<<<1, 32, 0, stream>>>(accs, 8);

  k_softmax<<<1000, 256, 0, stream>>>(C1, S1, 8000);
  k_softmax<<<1000, 256, 0, stream>>>(C2, S2, 8000);
  k_softmax<<<1000, 256, 0, stream>>>(Cg, Sg, 8000);
  k_softmax<<<2500, 256, 0, stream>>>(Cr, Sr, 20000);

  k_gemm_small<<<8000, 128, 0, stream>>>(S1, Agm, T1);
  k_gemm_small<<<8000, 128, 0, stream>>>(S2, Arm, T2);

  k_loss_rec<<<dim3(100, 125), 160, 0, stream>>>(G, T1, Sg, b_g, b_1, 8000, accs + 0);
  k_loss_rec<<<dim3(250, 125), 160, 0, stream>>>(R, T2, Sr, b_r, b_2, 20000, accs + 1);

  k_loss4<<<250, 224, 0, stream>>>(A, Sr, Sg, 20000, accs + 2);
  k_loss3<<<1, 256, 0, stream>>>(Agm, Arm, accs + 5);

  k_final<<<1, 32, 0, stream>>>(accs, alpha, out);

  (void)in_sizes; (void)n_in; (void)out_size; (void)ws_size;
}